// ColoredMLP_1451698946130
// MI455X (gfx1250) — compile-verified
//
#include <hip/hip_runtime.h>
#include <hip/hip_bf16.h>

typedef float v2f __attribute__((ext_vector_type(2)));
typedef float v4f __attribute__((ext_vector_type(4)));
typedef float v8f __attribute__((ext_vector_type(8)));

#define GDIM 64
#define FDIM 128
#define NCOL 4
#define ROWS_PER_BLK 128   // 8 row-tiles of 16
#define LN2F_   0.69314718055994530942f
#define LOG2EF_ 1.44269504088896340736f

// ---------------- binning (two-level atomics) ----------------

__global__ void zero_counters(int* cnt) {
    if (threadIdx.x < 64) cnt[threadIdx.x] = 0;
}

__global__ void count_colors(const int* __restrict__ colors, int* __restrict__ cnt, int E) {
    __shared__ int lc[NCOL];
    if (threadIdx.x < NCOL) lc[threadIdx.x] = 0;
    __syncthreads();
    int e = blockIdx.x * blockDim.x + threadIdx.x;
    if (e < E) atomicAdd(&lc[colors[e]], 1);
    __syncthreads();
    if (threadIdx.x < NCOL && lc[threadIdx.x] != 0)
        atomicAdd(&cnt[threadIdx.x], lc[threadIdx.x]);
}

__global__ void scatter_colors(const int* __restrict__ colors, int* __restrict__ cnt,
                               int* __restrict__ order, int E) {
    __shared__ int lc[NCOL];     // intra-block rank counters
    __shared__ int lbase[NCOL];  // this block's base within each color segment
    if (threadIdx.x < NCOL) lc[threadIdx.x] = 0;
    __syncthreads();
    int e = blockIdx.x * blockDim.x + threadIdx.x;
    int c = 0, myRank = 0;
    if (e < E) {
        c = colors[e];
        myRank = atomicAdd(&lc[c], 1);
    }
    __syncthreads();
    if (threadIdx.x < NCOL)
        lbase[threadIdx.x] = atomicAdd(&cnt[NCOL + threadIdx.x], lc[threadIdx.x]);
    __syncthreads();
    if (e < E) {
        int base = 0;
#pragma unroll
        for (int j = 0; j < NCOL; ++j) base += (j < c) ? cnt[j] : 0;
        order[base + lbase[c] + myRank] = e;
    }
}

// ---------------- fused 2-layer MLP GEMM ----------------

__device__ __forceinline__ float shifted_softplus(float x) {
    // softplus(x) - ln2, stable, raw hw transcendentals:
    //   max(x,0) + ln2 * (log2(1 + 2^(-|x|*log2e)) - 1)
    float u = __builtin_amdgcn_exp2f(-fabsf(x) * LOG2EF_);   // = exp(-|x|), in (0,1]
    float l = __builtin_amdgcn_logf(1.0f + u);               // log2(1+u), arg in (1,2]
    return fmaf(LN2F_, l - 1.0f, fmaxf(x, 0.0f));
}

__global__ void __launch_bounds__(256)
colored_mlp_gemm(const float* __restrict__ ea,     // [E,64]
                 const float* __restrict__ W1,     // [4,64,128]
                 const float* __restrict__ b1,     // [4,128]
                 const float* __restrict__ W2,     // [4,128,128]
                 const float* __restrict__ b2,     // [4,128]
                 const int*   __restrict__ cnt,    // [4] final counts
                 const int*   __restrict__ order,  // [E] color-sorted edge ids
                 float* __restrict__ out)          // [E,128]
{
    // map block -> (color, chunk)
    int ns[NCOL] = {cnt[0], cnt[1], cnt[2], cnt[3]};
    int b = blockIdx.x;
    int color = -1, segStart = 0, segLen = 0, chunk = 0;
    int s = 0;
#pragma unroll
    for (int c = 0; c < NCOL; ++c) {
        int t = (ns[c] + ROWS_PER_BLK - 1) / ROWS_PER_BLK;
        if (color < 0) {
            if (b < t) { color = c; chunk = b; segStart = s; segLen = ns[c]; }
            else b -= t;
        }
        s += ns[c];
    }
    if (color < 0) return;  // uniform early-out, before any WMMA

    const int tid  = threadIdx.x;
    const int w    = tid >> 5;        // wave id 0..7 -> column tile
    const int lane = tid & 31;
    const int nl   = lane & 15;       // N (or M) index within tile
    const int kh   = lane >> 4;       // K-half select

    const float* W1c = W1 + (size_t)color * GDIM * FDIM;
    const float* W2c = W2 + (size_t)color * FDIM * FDIM;
    const int ncol = w * 16 + nl;     // this wave+lane's output column

    // preload weight fragments into registers (reused across 8 row-tiles)
    v2f w1f[16];
#pragma unroll
    for (int kk = 0; kk < 16; ++kk) {
        int k0 = 4 * kk + 2 * kh;
        w1f[kk].x = W1c[(size_t)k0 * FDIM + ncol];
        w1f[kk].y = W1c[(size_t)(k0 + 1) * FDIM + ncol];
    }
    v2f w2f[32];
#pragma unroll
    for (int kk = 0; kk < 32; ++kk) {
        int k0 = 4 * kk + 2 * kh;
        w2f[kk].x = W2c[(size_t)k0 * FDIM + ncol];
        w2f[kk].y = W2c[(size_t)(k0 + 1) * FDIM + ncol];
    }
    const float bias1 = b1[color * FDIM + ncol];
    const float bias2 = b2[color * FDIM + ncol];

    // padded LDS tiles (conflict-free strides)
    __shared__ float As[16 * 68];     // 16 rows x 64 cols, stride 68
    __shared__ float Hs[16 * 132];    // 16 rows x 128 cols, stride 132

    const int rowBase = chunk * ROWS_PER_BLK;

    for (int rt = 0; rt < ROWS_PER_BLK / 16; ++rt) {
        const int tileRow = rowBase + rt * 16;

        // cooperative gather of A tile (streaming: non-temporal)
        {
            int r  = tid >> 4;          // 0..15
            int c4 = (tid & 15) * 4;    // 0..60
            int g  = tileRow + r;
            int gsel = (g < segLen) ? g : 0;           // clamp to a valid row
            int gi = order[segStart + gsel];
            v4f v = __builtin_nontemporal_load(
                        (const v4f*)&ea[(size_t)gi * GDIM + c4]);
            *(v4f*)&As[r * 68 + c4] = v;
        }
        __syncthreads();

        // ---- layer 1: 16x64 @ 64x16 with f32 WMMA 16x16x4 ----
        v8f acc = {};
#pragma unroll
        for (int kk = 0; kk < 16; ++kk) {
            int k0 = 4 * kk + 2 * kh;
            v2f af = *(const v2f*)&As[nl * 68 + k0];
            acc = __builtin_amdgcn_wmma_f32_16x16x4_f32(
                false, af, false, w1f[kk], (short)0, acc, false, false);
        }

        // bias + shifted softplus -> Hs
#pragma unroll
        for (int i = 0; i < 8; ++i) {
            int m = i + 8 * kh;
            Hs[m * 132 + ncol] = shifted_softplus(acc[i] + bias1);
        }
        __syncthreads();

        // ---- layer 2: 16x128 @ 128x16 ----
        v8f acc2 = {};
#pragma unroll
        for (int kk = 0; kk < 32; ++kk) {
            int k0 = 4 * kk + 2 * kh;
            v2f hf = *(const v2f*)&Hs[nl * 132 + k0];
            acc2 = __builtin_amdgcn_wmma_f32_16x16x4_f32(
                false, hf, false, w2f[kk], (short)0, acc2, false, false);
        }

        // scatter results, non-temporal (mask only the stores; WMMA already done)
#pragma unroll
        for (int i = 0; i < 8; ++i) {
            int m = i + 8 * kh;
            int g = tileRow + m;
            if (g < segLen) {
                int gi = order[segStart + g];
                __builtin_nontemporal_store(acc2[i] + bias2,
                                            &out[(size_t)gi * FDIM + ncol]);
            }
        }
        __syncthreads();   // before next iteration overwrites As/Hs
    }
}

// ---------------- launch ----------------

extern "C" void kernel_launch(void* const* d_in, const int* in_sizes, int n_in,
                              void* d_out, int out_size, void* d_ws, size_t ws_size,
                              hipStream_t stream) {
    const float* ea     = (const float*)d_in[0];
    const int*   colors = (const int*)d_in[1];
    const float* W1     = (const float*)d_in[2];
    const float* b1     = (const float*)d_in[3];
    const float* W2     = (const float*)d_in[4];
    const float* b2     = (const float*)d_in[5];
    float* out = (float*)d_out;

    const int E = in_sizes[1];          // colors is [E]

    int* cnt   = (int*)d_ws;                        // [0..3] counts, [4..7] cursors
    int* order = (int*)((char*)d_ws + 256);         // E ints

    zero_counters<<<1, 64, 0, stream>>>(cnt);
    int blk = 256;
    int grd = (E + blk - 1) / blk;
    count_colors<<<grd, blk, 0, stream>>>(colors, cnt, E);
    scatter_colors<<<grd, blk, 0, stream>>>(colors, cnt, order, E);

    int gemm_blocks = (E + ROWS_PER_BLK - 1) / ROWS_PER_BLK + NCOL;
    colored_mlp_gemm<<<gemm_blocks, 256, 0, stream>>>(
        ea, W1, b1, W2, b2, cnt, order, out);
}